// TemporalSpikeCoder_78125455114738
// MI455X (gfx1250) — compile-verified
//
#include <hip/hip_runtime.h>
#include <hip/hip_bf16.h>

// TemporalSpikeCoder latency encoding, MI455X (gfx1250).
//
// out[b,t,f] = 1.0f if int((1-x[b,f])*100) == t else 0.0f
//   x:   [2048, 1024] f32
//   out: [2048, 100, 1024] f32  (209,715,200 elements, 839 MB)
//
// Pure store-bandwidth problem (~36 us floor at 23.3 TB/s). No contraction ->
// WMMA inapplicable. Strategy: one thread per float4 of x; 100 fully-unrolled
// non-temporal global_store_b128 zero-fills down the t-column (512 B per wave
// per store, NT so 839 MB of streaming output doesn't thrash the 192 MB L2),
// then <=4 scalar 1.0f overwrites at the spike positions. Same-wave stores
// complete in order (CDNA5 ISA VMEM ordering), so the overwrite is safe.

#define BATCH 2048
#define FEAT  1024
#define NSTEP 100

typedef __attribute__((ext_vector_type(4))) float v4f;

__global__ __launch_bounds__(256) void spike_latency_encode(
    const float* __restrict__ x, float* __restrict__ out)
{
    const int gid  = blockIdx.x * blockDim.x + threadIdx.x;   // one thread per 4 features
    const int b    = gid >> 8;                                // FEAT/4 == 256
    const int f    = (gid & 255) << 2;

    // Load 4 features, compute spike times exactly like jnp:
    // f32 (1.0 - x), f32 * 100, truncate to int32. Sub-then-mul shape means
    // no fma contraction can perturb the result.
    const v4f xv = *reinterpret_cast<const v4f*>(x + (size_t)b * FEAT + f);
    const int st0 = (int)((1.0f - xv.x) * 100.0f);
    const int st1 = (int)((1.0f - xv.y) * 100.0f);
    const int st2 = (int)((1.0f - xv.z) * 100.0f);
    const int st3 = (int)((1.0f - xv.w) * 100.0f);

    float* col = out + (size_t)b * (NSTEP * FEAT) + f;

    // Phase 1: stream zeros down the whole t-column. Fully unrolled: one base
    // address, 100 x global_store_b128 with immediate offsets (t*4096 B fits
    // the signed 24-bit IOFFSET), non-temporal so they rinse through L2.
    const v4f z = (v4f)(0.0f);
#pragma unroll
    for (int t = 0; t < NSTEP; ++t) {
        __builtin_nontemporal_store(z, reinterpret_cast<v4f*>(col + (size_t)t * FEAT));
    }

    // Phase 2: drop the ones. st == NSTEP (x == 0) spikes never, matching the
    // reference's masked one-hot. Unsigned compare also guards any st < 0.
    if ((unsigned)st0 < NSTEP) col[(size_t)st0 * FEAT + 0] = 1.0f;
    if ((unsigned)st1 < NSTEP) col[(size_t)st1 * FEAT + 1] = 1.0f;
    if ((unsigned)st2 < NSTEP) col[(size_t)st2 * FEAT + 2] = 1.0f;
    if ((unsigned)st3 < NSTEP) col[(size_t)st3 * FEAT + 3] = 1.0f;
}

extern "C" void kernel_launch(void* const* d_in, const int* in_sizes, int n_in,
                              void* d_out, int out_size, void* d_ws, size_t ws_size,
                              hipStream_t stream)
{
    (void)in_sizes; (void)n_in; (void)out_size; (void)d_ws; (void)ws_size;
    const float* x = (const float*)d_in[0];
    float* out     = (float*)d_out;

    const int threads = 256;                       // 8 wave32s per block
    const int total   = BATCH * (FEAT / 4);        // 524,288 threads
    const int blocks  = total / threads;           // 2048 blocks
    spike_latency_encode<<<blocks, threads, 0, stream>>>(x, out);
}